// GNN_Multi_Layer_9895604650407
// MI455X (gfx1250) — compile-verified
//
#include <hip/hip_runtime.h>
#include <hip/hip_bf16.h>
#include <stdint.h>

// ---------------------------------------------------------------------------
// 2-layer GCN for MI455X (gfx1250, wave32).
// Memory-bound (edge gather/scatter ~1.3GB >> 2.5 GFLOP of GEMM), so GEMMs use
// full-precision V_WMMA_F32_16X16X4_F32 and scatters use native f32 atomics.
// ---------------------------------------------------------------------------

typedef __attribute__((ext_vector_type(2))) float v2f;
typedef __attribute__((ext_vector_type(8))) float v8f;

#define N_CH1 128
#define N_CH2 64

__device__ __forceinline__ void atomicAddF32(float* p, float v) {
  __hip_atomic_fetch_add(p, v, __ATOMIC_RELAXED, __HIP_MEMORY_SCOPE_AGENT);
}

// ---- row-wise softmax over 128 channels: one wave32 per node, 4 ch/lane ----
__global__ void softmax_rows_kernel(const float* __restrict__ X,
                                    float* __restrict__ H, int n) {
  int wave = blockIdx.x * (blockDim.x >> 5) + (threadIdx.x >> 5);
  int lane = threadIdx.x & 31;
  if (wave >= n) return;
  const float4* xr = (const float4*)(X + (size_t)wave * N_CH1);
  float4 v = xr[lane];
  float m = fmaxf(fmaxf(v.x, v.y), fmaxf(v.z, v.w));
#pragma unroll
  for (int off = 16; off > 0; off >>= 1) m = fmaxf(m, __shfl_xor(m, off, 32));
  float e0 = __expf(v.x - m), e1 = __expf(v.y - m);
  float e2 = __expf(v.z - m), e3 = __expf(v.w - m);
  float s = e0 + e1 + e2 + e3;
#pragma unroll
  for (int off = 16; off > 0; off >>= 1) s += __shfl_xor(s, off, 32);
  float inv = 1.0f / s;
  float4 o = make_float4(e0 * inv, e1 * inv, e2 * inv, e3 * inv);
  ((float4*)(H + (size_t)wave * N_CH1))[lane] = o;
}

// ---- degree handling ----
__global__ void deg_init_kernel(float* __restrict__ deg, int n) {
  int i = blockIdx.x * blockDim.x + threadIdx.x;
  if (i < n) deg[i] = 1.0f;  // self-loop weight
}

__global__ void deg_scatter_kernel(const long long* __restrict__ col,
                                   const float* __restrict__ ew,
                                   float* __restrict__ deg, int E) {
  int e = blockIdx.x * blockDim.x + threadIdx.x;
  if (e < E) atomicAddF32(&deg[col[e]], ew[e]);
}

__global__ void dinv_kernel(const float* __restrict__ deg,
                            float* __restrict__ dinv, int n) {
  int i = blockIdx.x * blockDim.x + threadIdx.x;
  if (i < n) {
    float d = deg[i];
    dinv[i] = (d > 0.0f) ? rsqrtf(d) : 0.0f;
  }
}

// ---- WMMA fp32 GEMM: C[M,N] = A[M,K] @ W[K,N], one wave per 16x16 tile ----
// ISA layout (32-bit, 16x16x4): A-frag reg r, lane-half h => K=k0+r+2h, M=lane%16.
// B-frag reg r => row K=k0+r+2h, N=lane%16. C/D reg r => row r+8h, col lane%16.
__global__ void gemm_wmma_f32_kernel(const float* __restrict__ A,
                                     const float* __restrict__ W,
                                     float* __restrict__ C,
                                     int M, int K, int N) {
  const int lane  = threadIdx.x & 31;
  const int gwave = blockIdx.x * (blockDim.x >> 5) + (threadIdx.x >> 5);
  const int ntiles = N >> 4;
  const int mt = gwave / ntiles;
  const int nt = gwave - mt * ntiles;
  if (mt * 16 >= M) return;  // wave-uniform exit; EXEC all-ones for WMMA below
  const int half = lane >> 4;
  const int lm   = lane & 15;
  const int mrow = mt * 16 + lm;
  const int ncol = nt * 16 + lm;
  const float* arow = A + (size_t)mrow * K;
  v8f acc = {};
  for (int k0 = 0; k0 < K; k0 += 4) {
    const int ka = k0 + 2 * half;
    float2 av = *(const float2*)(arow + ka);  // contiguous K pair
    v2f a, b;
    a.x = av.x; a.y = av.y;
    b.x = W[(size_t)ka * N + ncol];
    b.y = W[(size_t)(ka + 1) * N + ncol];
    acc = __builtin_amdgcn_wmma_f32_16x16x4_f32(
        /*neg_a=*/false, a, /*neg_b=*/false, b,
        /*c_mod=*/(short)0, acc, /*reuse_a=*/false, /*reuse_b=*/false);
  }
#pragma unroll
  for (int r = 0; r < 8; ++r)
    C[(size_t)(mt * 16 + r + 8 * half) * N + ncol] = acc[r];
}

// ---- aggregation: init Out with self-loop term dinv[i]^2 * Hin[i] ----
__global__ void agg_self_init_kernel(const float* __restrict__ Hin,
                                     const float* __restrict__ dinv,
                                     float* __restrict__ Out, int n, int C) {
  int idx = blockIdx.x * blockDim.x + threadIdx.x;
  int total = n * C;
  if (idx >= total) return;
  int i = idx / C;
  float d = dinv[i];
  Out[idx] = d * d * Hin[idx];
}

// ---- per-edge scatter: one wave per edge, PER channels per lane ----
template <int CH, int PER>
__global__ void edge_scatter_kernel(const long long* __restrict__ row,
                                    const long long* __restrict__ col,
                                    const float* __restrict__ ew,
                                    const float* __restrict__ dinv,
                                    const float* __restrict__ Hin,
                                    float* __restrict__ Out, int E) {
  int wave = blockIdx.x * (blockDim.x >> 5) + (threadIdx.x >> 5);
  int lane = threadIdx.x & 31;
  if (wave >= E) return;
  long long r = row[wave];
  long long c = col[wave];
  float norm = dinv[r] * ew[wave] * dinv[c];
  const float* src = Hin + (size_t)r * CH + lane * PER;
  float* dst = Out + (size_t)c * CH + lane * PER;
  float vals[PER];
#pragma unroll
  for (int j = 0; j < PER; ++j) vals[j] = src[j];  // coalesced gather (L2-hot)
#pragma unroll
  for (int j = 0; j < PER; ++j) atomicAddF32(&dst[j], norm * vals[j]);
}

// ---- bias (+ optional ReLU) ----
template <bool RELU>
__global__ void bias_act_kernel(float* __restrict__ H,
                                const float* __restrict__ b, int n, int C) {
  int idx = blockIdx.x * blockDim.x + threadIdx.x;
  int total = n * C;
  if (idx >= total) return;
  int c = idx - (idx / C) * C;
  float v = H[idx] + b[c];
  H[idx] = RELU ? fmaxf(v, 0.0f) : v;
}

extern "C" void kernel_launch(void* const* d_in, const int* in_sizes, int n_in,
                              void* d_out, int out_size, void* d_ws, size_t ws_size,
                              hipStream_t stream) {
  const float*     x   = (const float*)d_in[0];
  const long long* ei  = (const long long*)d_in[1];
  const float*     ew  = (const float*)d_in[2];
  const float*     W1  = (const float*)d_in[3];
  const float*     b1  = (const float*)d_in[4];
  const float*     W2  = (const float*)d_in[5];
  const float*     b2  = (const float*)d_in[6];
  float*           out = (float*)d_out;

  const int n = in_sizes[0] / N_CH1;    // 50000
  const int E = in_sizes[1] / 2;        // 800000
  const long long* row = ei;
  const long long* col = ei + E;

  // workspace layout
  char* ws = (char*)d_ws;
  const size_t featBytes = (size_t)n * N_CH1 * sizeof(float);
  float* bufA = (float*)ws;                         // softmax h, then agg1 out
  float* bufB = (float*)(ws + featBytes);           // h1, then h3
  float* deg  = (float*)(ws + 2 * featBytes);
  float* dinv = (float*)(ws + 2 * featBytes + (size_t)n * sizeof(float));

  const int TB = 256;
  const int wavesPerBlock = TB / 32;

  // 1) h = softmax(x) -> bufA
  softmax_rows_kernel<<<(n + wavesPerBlock - 1) / wavesPerBlock, TB, 0, stream>>>(
      x, bufA, n);

  // 2) degree + dinv
  deg_init_kernel<<<(n + TB - 1) / TB, TB, 0, stream>>>(deg, n);
  deg_scatter_kernel<<<(E + TB - 1) / TB, TB, 0, stream>>>(col, ew, deg, E);
  dinv_kernel<<<(n + TB - 1) / TB, TB, 0, stream>>>(deg, dinv, n);

  // 3) h1 = h @ W1 -> bufB   (3125*8 tiles)
  {
    int tiles = (n / 16) * (N_CH1 / 16);
    gemm_wmma_f32_kernel<<<(tiles + wavesPerBlock - 1) / wavesPerBlock, TB, 0, stream>>>(
        bufA, W1, bufB, n, N_CH1, N_CH1);
  }

  // 4) agg1: bufA = dinv^2*h1 (self) + scatter(norm*h1[row] -> col); relu(+b1)
  agg_self_init_kernel<<<((n * N_CH1) + TB - 1) / TB, TB, 0, stream>>>(
      bufB, dinv, bufA, n, N_CH1);
  edge_scatter_kernel<N_CH1, 4><<<(E + wavesPerBlock - 1) / wavesPerBlock, TB, 0, stream>>>(
      row, col, ew, dinv, bufB, bufA, E);
  bias_act_kernel<true><<<((n * N_CH1) + TB - 1) / TB, TB, 0, stream>>>(
      bufA, b1, n, N_CH1);

  // 5) h3 = h2 @ W2 -> bufB (reuse; 50000x64 fits)
  {
    int tiles = (n / 16) * (N_CH2 / 16);
    gemm_wmma_f32_kernel<<<(tiles + wavesPerBlock - 1) / wavesPerBlock, TB, 0, stream>>>(
        bufA, W2, bufB, n, N_CH1, N_CH2);
  }

  // 6) agg2 into d_out + bias
  agg_self_init_kernel<<<((n * N_CH2) + TB - 1) / TB, TB, 0, stream>>>(
      bufB, dinv, out, n, N_CH2);
  edge_scatter_kernel<N_CH2, 2><<<(E + wavesPerBlock - 1) / wavesPerBlock, TB, 0, stream>>>(
      row, col, ew, dinv, bufB, out, E);
  bias_act_kernel<false><<<((n * N_CH2) + TB - 1) / TB, TB, 0, stream>>>(
      out, b2, n, N_CH2);
}